// RouterSinkhorn_17532056502442
// MI455X (gfx1250) — compile-verified
//
#include <hip/hip_runtime.h>
#include <hip/hip_bf16.h>
#include <math.h>

// ---------------------------------------------------------------------------
// Router + Sinkhorn (MoE top-1 routing) for MI455X / gfx1250, wave32.
//   T = 8192 tokens (4096 seq x 2 batch), H = 4096 hidden, E = 64 experts.
// d_out layout (float):  [0, T*E)       router_logits
//                        [T*E, 2T*E)    sigmoid affinities
//                        [2T*E, 2T*E+T) expert index (as float 0..63)
// d_ws layout (float):   [0, T)  d0     [T, T+E)  d1
// ---------------------------------------------------------------------------

#define T_TOK 8192
#define HDIM  4096
#define NEXP  64
#define KC    64      // K chunk staged in LDS per block
#define LDA   66      // even pad: row base 8B-aligned -> ds_load_b64 frags
#define LDW   66
#define EPS   1e-8f

typedef float v2f __attribute__((ext_vector_type(2)));
typedef float v8f __attribute__((ext_vector_type(8)));

// ---------------------------------------------------------------------------
// GEMM: logits = hidden(T,H) @ W(H,E); also emits sigmoid(logits).
// 512 blocks x 128 threads (4 waves). Block -> 16 rows x 64 cols.
// Wave w -> 16x16 tile at columns [16w, 16w+16).
// A tile (16xKC) and W chunk (transposed, NEXPxKC) staged in LDS so every
// WMMA operand is one aligned ds_load_b64. V_WMMA_F32_16X16X4_F32 over K.
// ---------------------------------------------------------------------------
__global__ __launch_bounds__(128)
void router_gemm_wmma(const float* __restrict__ A,   // (T,H)
                      const float* __restrict__ W,   // (H,E)
                      float* __restrict__ logits,    // (T,E)
                      float* __restrict__ aff)       // (T,E)
{
    __shared__ __align__(16) float lds_a[16 * LDA];    // A tile  [row][k]
    __shared__ __align__(16) float lds_w[NEXP * LDW];  // W chunk [e][k] (transposed)

    const int row0 = blockIdx.x << 4;          // 16 rows per block
    const int wave = threadIdx.x >> 5;
    const int lane = threadIdx.x & 31;
    const int n0   = wave << 4;                // 16 cols per wave
    const int lrow = lane & 15;
    const int hi   = (lane >> 4) << 1;         // 0: K pair {0,1}, 2: K pair {2,3}

    v8f acc = {};

    for (int kc = 0; kc < HDIM; kc += KC) {
        // --- cooperative A-tile load: 16 rows x KC, coalesced ---
        for (int i = threadIdx.x; i < 16 * KC; i += 128) {
            const int r = i >> 6;              // /KC
            const int c = i & (KC - 1);
            lds_a[r * LDA + c] = A[(size_t)(row0 + r) * HDIM + kc + c];
        }
        // --- cooperative W-chunk load, transposed to [e][k] ---
        // global read coalesced over e; LDS write bank = (2e+k)%64: conflict-free
        for (int i = threadIdx.x; i < KC * NEXP; i += 128) {
            const int k = i >> 6;
            const int e = i & (NEXP - 1);
            lds_w[e * LDW + k] = W[(size_t)(kc + k) * NEXP + e];
        }
        __syncthreads();

        // prefetch next A chunk into cache while we compute (global_prefetch_b8)
        if (kc + KC < HDIM) {
            const float* nxt = &A[(size_t)(row0 + (threadIdx.x >> 3)) * HDIM
                                  + kc + KC + ((threadIdx.x & 7) << 3)];
            __builtin_prefetch(nxt, 0, 0);
        }

        #pragma unroll
        for (int kk = 0; kk < KC; kk += 4) {
            // A frag (16x4 f32): lane<16 -> K=kk,kk+1 ; lane>=16 -> K=kk+2,kk+3
            const v2f a = *(const v2f*)&lds_a[lrow * LDA + kk + hi];
            // B frag (4x16 f32), mirrored striping, from transposed W chunk
            const v2f b = *(const v2f*)&lds_w[(n0 + lrow) * LDW + kk + hi];
            acc = __builtin_amdgcn_wmma_f32_16x16x4_f32(
                      /*neg_a=*/false, a, /*neg_b=*/false, b,
                      /*c_mod=*/(short)0, acc,
                      /*reuse_a=*/false, /*reuse_b=*/false);
        }
        __syncthreads();
    }

    // D layout: lane<16 -> (M=i, N=lane); lane>=16 -> (M=8+i, N=lane-16)
    #pragma unroll
    for (int i = 0; i < 8; ++i) {
        const int m = (hi ? 8 : 0) + i;
        const size_t idx = (size_t)(row0 + m) * NEXP + n0 + lrow;
        const float x = acc[i];
        logits[idx] = x;
        aff[idx]    = 1.0f / (1.0f + __expf(-x));
    }
}

// --------------------------- Sinkhorn kernels ------------------------------
__global__ __launch_bounds__(64)
void sink_init(float* __restrict__ d1)
{
    d1[threadIdx.x] = 1.0f;
}

// d0[t] = (1/T) / (sum_e d1[e]*exp(logits[t,e]) + eps)   -- one thread per row
__global__ __launch_bounds__(256)
void sink_d0(const float* __restrict__ logits, const float* __restrict__ d1,
             float* __restrict__ d0)
{
    __shared__ float s_d1[NEXP];
    if (threadIdx.x < NEXP) s_d1[threadIdx.x] = d1[threadIdx.x];
    __syncthreads();

    const int t = blockIdx.x * 256 + threadIdx.x;
    const float4* row = (const float4*)&logits[(size_t)t * NEXP];
    float sum = 0.0f;
    #pragma unroll
    for (int q = 0; q < NEXP / 4; ++q) {
        const float4 v = row[q];
        sum += s_d1[4*q + 0] * __expf(v.x) + s_d1[4*q + 1] * __expf(v.y)
             + s_d1[4*q + 2] * __expf(v.z) + s_d1[4*q + 3] * __expf(v.w);
    }
    d0[t] = (1.0f / (float)T_TOK) / (sum + EPS);
}

// d1[e] = (1/E) / (sum_t d0[t]*exp(logits[t,e]) + eps)  -- one block per expert
__global__ __launch_bounds__(256)
void sink_d1(const float* __restrict__ logits, const float* __restrict__ d0,
             float* __restrict__ d1)
{
    __shared__ float red[256];
    const int e = blockIdx.x;
    float sum = 0.0f;
    for (int t = threadIdx.x; t < T_TOK; t += 256)
        sum += d0[t] * __expf(logits[(size_t)t * NEXP + e]);
    red[threadIdx.x] = sum;
    __syncthreads();
    #pragma unroll
    for (int s = 128; s > 0; s >>= 1) {
        if (threadIdx.x < s) red[threadIdx.x] += red[threadIdx.x + s];
        __syncthreads();
    }
    if (threadIdx.x == 0) d1[e] = (1.0f / (float)NEXP) / (red[0] + EPS);
}

// argmax_e d1[e]*C[t,e]*d0[t] == argmax_e d1[e]*exp(logits[t,e])  (d0[t] > 0)
__global__ __launch_bounds__(256)
void sink_argmax(const float* __restrict__ logits, const float* __restrict__ d1,
                 float* __restrict__ out_idx)
{
    __shared__ float s_d1[NEXP];
    if (threadIdx.x < NEXP) s_d1[threadIdx.x] = d1[threadIdx.x];
    __syncthreads();

    const int t = blockIdx.x * 256 + threadIdx.x;
    const float4* row = (const float4*)&logits[(size_t)t * NEXP];
    float best = -1.0f;
    int   bi   = 0;
    #pragma unroll
    for (int q = 0; q < NEXP / 4; ++q) {
        const float4 v = row[q];
        float c0 = s_d1[4*q + 0] * __expf(v.x);
        float c1 = s_d1[4*q + 1] * __expf(v.y);
        float c2 = s_d1[4*q + 2] * __expf(v.z);
        float c3 = s_d1[4*q + 3] * __expf(v.w);
        if (c0 > best) { best = c0; bi = 4*q + 0; }
        if (c1 > best) { best = c1; bi = 4*q + 1; }
        if (c2 > best) { best = c2; bi = 4*q + 2; }
        if (c3 > best) { best = c3; bi = 4*q + 3; }
    }
    out_idx[t] = (float)bi;   // indices 0..63 are exact in f32
}

// ---------------------------------------------------------------------------
extern "C" void kernel_launch(void* const* d_in, const int* in_sizes, int n_in,
                              void* d_out, int out_size, void* d_ws, size_t ws_size,
                              hipStream_t stream)
{
    (void)in_sizes; (void)n_in; (void)out_size; (void)ws_size;

    const float* hidden = (const float*)d_in[0];   // (4096,2,4096) = (T,H) flat
    const float* W      = (const float*)d_in[1];   // (H,E)

    float* logits = (float*)d_out;                         // T*E
    float* aff    = logits + (size_t)T_TOK * NEXP;         // T*E
    float* oidx   = aff    + (size_t)T_TOK * NEXP;         // T

    float* d0 = (float*)d_ws;                              // T
    float* d1 = d0 + T_TOK;                                // E

    // 1) router GEMM (WMMA f32) + sigmoid affinities
    router_gemm_wmma<<<T_TOK / 16, 128, 0, stream>>>(hidden, W, logits, aff);

    // 2) Sinkhorn: 30 iterations, split-kernel global sync
    sink_init<<<1, 64, 0, stream>>>(d1);
    for (int it = 0; it < 30; ++it) {
        sink_d0<<<T_TOK / 256, 256, 0, stream>>>(logits, d1, d0);
        sink_d1<<<NEXP, 256, 0, stream>>>(logits, d0, d1);
    }

    // 3) top-1 expert index
    sink_argmax<<<T_TOK / 256, 256, 0, stream>>>(logits, d1, oidx);
}